// CorticalLayer_65369402245526
// MI455X (gfx1250) — compile-verified
//
#include <hip/hip_runtime.h>
#include <hip/hip_bf16.h>
#include <math.h>

#define N_NEU 8192
#define B_BATCH 32
#define E_EDGE 1048576
#define G_GRP 64

#define KSPLIT 8
#define KT 32     // k per double-buffered chunk
#define LDP 36    // LDS row stride (floats): 36*m mod 64 distinct -> conflict-free

typedef __attribute__((ext_vector_type(2))) float v2f;
typedef __attribute__((ext_vector_type(8))) float v8f;

// ---------------------------------------------------------------------------
// prep: column mean of x_in, transpose x_in -> xT (N,32), stdp_mod[n] (n < N)
// ---------------------------------------------------------------------------
__global__ void prep_kernel(const float* __restrict__ x_in,
                            const float* __restrict__ prev_spk,
                            const float* __restrict__ delay_buf,
                            const int* __restrict__ delays,
                            float* __restrict__ xT,
                            float* __restrict__ stdp_mod) {
    int n = blockIdx.x * blockDim.x + threadIdx.x;
    if (n >= N_NEU) return;
    float sum = 0.f;
#pragma unroll
    for (int b = 0; b < B_BATCH; ++b) {
        float v = x_in[(size_t)b * N_NEU + n];   // coalesced across n
        sum += v;
        xT[(size_t)n * B_BATCH + b] = v;
    }
    float xmean = sum * (1.0f / B_BATCH);
    float ps = prev_spk[n];
    float delayed = delay_buf[(size_t)delays[n] * E_EDGE + n];
    stdp_mod[n] = 0.01f * delayed * ps - 0.012f * xmean * ps;
}

// ---------------------------------------------------------------------------
// group sums of prev_spk: 64 groups x 128
// ---------------------------------------------------------------------------
__global__ void gsum_kernel(const float* __restrict__ prev_spk,
                            float* __restrict__ gsum) {
    __shared__ float red[128];
    int t = threadIdx.x;
    red[t] = prev_spk[blockIdx.x * 128 + t];
    __syncthreads();
    for (int s = 64; s > 0; s >>= 1) {
        if (t < s) red[t] += red[t + s];
        __syncthreads();
    }
    if (t == 0) gsum[blockIdx.x] = red[0];
}

__global__ void zero_kernel(float* __restrict__ p, int n) {
    int i = blockIdx.x * blockDim.x + threadIdx.x;
    if (i < n) p[i] = 0.f;
}

// ---------------------------------------------------------------------------
// scatter: wave-per-edge, lane = batch. Coalesced gather from xT (N,32),
// coalesced 128B atomic burst into synT (N,32). 1M waves.
// ---------------------------------------------------------------------------
__global__ void scatter_kernel(const float* __restrict__ xT,
                               const int* __restrict__ src,
                               const int* __restrict__ dst,
                               const float* __restrict__ w_hat,
                               float* __restrict__ synT) {
    unsigned gid = blockIdx.x * blockDim.x + threadIdx.x;
    unsigned e = gid >> 5;
    unsigned lane = gid & 31;
    if (e >= E_EDGE) return;
    int s = src[e];                 // lane-uniform -> HW broadcast
    int d = dst[e];
    float w = w_hat[e];
    float v = xT[(size_t)s * B_BATCH + lane] * w;
    atomicAdd(&synT[(size_t)d * B_BATCH + lane], v);
}

// ---------------------------------------------------------------------------
// GEMM: acc[b,n] += sum_k synT[k,b] * W[n,k]   (f32 WMMA 16x16x4)
// Block: 256 thr = 8 waves; block tile = 128 N-cols, K-split across blocks.
// Double-buffered LDS tiles filled by GLOBAL_LOAD_ASYNC_TO_LDS (ASYNCcnt),
// overlapping the HBM stream of W with WMMA compute.
// ---------------------------------------------------------------------------
__global__ void __launch_bounds__(256)
gemm_kernel(const float* __restrict__ W,
            const float* __restrict__ synT,
            float* __restrict__ acc) {
    __shared__ float Ws[2][128 * LDP];   // 2 x 18 KB
    __shared__ float As[2][32 * LDP];    // 2 x 4.5 KB

    int nblk = blockIdx.x & 63;          // 64 N-blocks of 128
    int kblk = blockIdx.x >> 6;          // KSPLIT k-chunks
    int n_base = nblk * 128;
    int k_begin = kblk * (N_NEU / KSPLIT);

    int wave = threadIdx.x >> 5;
    int lane = threadIdx.x & 31;
    int m_lane = lane & 15;
    int k_lane = (lane >> 4) << 1;       // 0 or 2 (ISA f32 A/B lane layout)
    int n_row = wave * 16 + m_lane;      // local N row for B fragment

    // issue async global->LDS DMA for one chunk (8 async ops per thread):
    //   W tile: 128 rows x 32 floats  (b128 per lane, coalesced rows)
    //   A tile: 32 b   x 32 k         (b32 per lane, transposing scatter)
    auto issue = [&](int k0, int buf) {
#pragma unroll
        for (int i = 0; i < 4; ++i) {
            int idx = (int)threadIdx.x + i * 256;     // 0..1023
            int r = idx >> 3;                         // row 0..127
            int c4 = idx & 7;                         // float4 col 0..7
            unsigned lds = (unsigned)(uintptr_t)&Ws[buf][r * LDP + (c4 << 2)];
            unsigned goff = ((unsigned)(n_base + r) * N_NEU +
                             (unsigned)k0 + (unsigned)(c4 << 2)) * 4u;
            asm volatile("global_load_async_to_lds_b128 %0, %1, %2"
                         :: "v"(lds), "v"(goff), "s"(W) : "memory");
        }
#pragma unroll
        for (int i = 0; i < 4; ++i) {
            int idx = (int)threadIdx.x + i * 256;     // 0..1023
            int bb = idx & 31;
            int kl = idx >> 5;
            unsigned lds = (unsigned)(uintptr_t)&As[buf][bb * LDP + kl];
            unsigned goff = ((unsigned)(k0 + kl) * B_BATCH + (unsigned)bb) * 4u;
            asm volatile("global_load_async_to_lds_b32 %0, %1, %2"
                         :: "v"(lds), "v"(goff), "s"(synT) : "memory");
        }
    };

    v8f c0 = {0.f, 0.f, 0.f, 0.f, 0.f, 0.f, 0.f, 0.f};
    v8f c1 = c0;

    const int nchunks = (N_NEU / KSPLIT) / KT;        // 32

    issue(k_begin, 0);                                 // prefetch chunk 0
    for (int c = 0; c < nchunks; ++c) {
        int buf = c & 1;
        if (c + 1 < nchunks) {
            issue(k_begin + (c + 1) * KT, buf ^ 1);    // prefetch next chunk
            // async ops complete in order: <=8 outstanding means chunk c done
            asm volatile("s_wait_asynccnt 0x8" ::: "memory");
        } else {
            asm volatile("s_wait_asynccnt 0x0" ::: "memory");
        }
        __syncthreads();                               // cross-wave visibility
#pragma unroll
        for (int kk = 0; kk < KT; kk += 4) {
            v2f a0 = *(const v2f*)&As[buf][m_lane * LDP + kk + k_lane];
            v2f a1 = *(const v2f*)&As[buf][(m_lane + 16) * LDP + kk + k_lane];
            v2f bf = *(const v2f*)&Ws[buf][n_row * LDP + kk + k_lane];
            c0 = __builtin_amdgcn_wmma_f32_16x16x4_f32(
                     false, a0, false, bf, (short)0, c0, false, false);
            c1 = __builtin_amdgcn_wmma_f32_16x16x4_f32(
                     false, a1, false, bf, (short)0, c1, false, false);
        }
        __syncthreads();   // all reads of buf done before it is refilled
    }

    // D layout: VGPR v -> M = v + (lane>=16 ? 8 : 0), N = lane&15
    int n_out = n_base + wave * 16 + m_lane;
    int m_hi = (lane >> 4) << 3;
#pragma unroll
    for (int v = 0; v < 8; ++v) {
        int m = v + m_hi;
        atomicAdd(&acc[(size_t)m * N_NEU + n_out], c0[v]);
        atomicAdd(&acc[(size_t)(m + 16) * N_NEU + n_out], c1[v]);
    }
}

// ---------------------------------------------------------------------------
// epilogue: g_slow scale, +0.1*stdp, mask, inhibition, leak, sigmoid
// ---------------------------------------------------------------------------
__global__ void epilogue_kernel(const float* __restrict__ acc,
                                const float* __restrict__ stdp_mod,
                                const float* __restrict__ gsum,
                                const float* __restrict__ prev_spk,
                                const float* __restrict__ mem,
                                const float* __restrict__ theta_ie,
                                const int* __restrict__ mask,
                                const float* __restrict__ g_slow,
                                float* __restrict__ out) {
    int gid = blockIdx.x * blockDim.x + threadIdx.x;
    if (gid >= B_BATCH * N_NEU) return;
    int n = gid & (N_NEU - 1);
    float gs = g_slow[0];
    float val = acc[gid] * gs + 0.1f * stdp_mod[n];
    val *= (float)mask[n];
    float inh = 0.5f * (gsum[n >> 7] - prev_spk[n]);
    float mem_new = 0.9f * mem[gid] + val - inh;
    float z = 10.0f * (mem_new - 1.0f - theta_ie[n]);
    out[gid] = 1.0f / (1.0f + expf(-z));
}

// ---------------------------------------------------------------------------
extern "C" void kernel_launch(void* const* d_in, const int* in_sizes, int n_in,
                              void* d_out, int out_size, void* d_ws, size_t ws_size,
                              hipStream_t stream) {
    const float* x_in      = (const float*)d_in[0];
    const float* w_hat     = (const float*)d_in[1];
    const float* W         = (const float*)d_in[2];
    const float* delay_buf = (const float*)d_in[3];
    const float* prev_spk  = (const float*)d_in[4];
    const float* mem       = (const float*)d_in[5];
    const float* theta_ie  = (const float*)d_in[6];
    const float* g_slow    = (const float*)d_in[7];
    const int*   hyper     = (const int*)d_in[8];   // [0..E)=src, [E..2E)=dst
    const int*   delays    = (const int*)d_in[9];
    const int*   mask      = (const int*)d_in[10];
    float* out = (float*)d_out;

    float* ws       = (float*)d_ws;
    float* xT       = ws;                              // N*32
    float* synT     = xT   + (size_t)N_NEU * B_BATCH;  // N*32
    float* acc      = synT + (size_t)N_NEU * B_BATCH;  // B*N
    float* stdp_mod = acc  + (size_t)B_BATCH * N_NEU;  // N
    float* gsum     = stdp_mod + N_NEU;                // 64

    // zero synT + acc (contiguous)
    int zcount = 2 * N_NEU * B_BATCH;
    zero_kernel<<<(zcount + 255) / 256, 256, 0, stream>>>(synT, zcount);

    prep_kernel<<<N_NEU / 256, 256, 0, stream>>>(x_in, prev_spk, delay_buf,
                                                 delays, xT, stdp_mod);
    gsum_kernel<<<G_GRP, 128, 0, stream>>>(prev_spk, gsum);

    scatter_kernel<<<(E_EDGE * 32) / 256, 256, 0, stream>>>(
        xT, hyper, hyper + E_EDGE, w_hat, synT);

    gemm_kernel<<<64 * KSPLIT, 256, 0, stream>>>(W, synT, acc);

    epilogue_kernel<<<(B_BATCH * N_NEU) / 256, 256, 0, stream>>>(
        acc, stdp_mod, gsum, prev_spk, mem, theta_ie, mask, g_slow, out);
}